// CETopKLoss_30081950941414
// MI455X (gfx1250) — compile-verified
//
#include <hip/hip_runtime.h>
#include <cstdint>
#include <cstddef>

#define BLOCK   256          // 8 wave32s per block
#define CHUNK   4096         // floats per staged chunk (16 KB)
#define GROUPS  4            // CHUNK / (BLOCK*4): quads per thread per chunk
#define BATCH_OPS 16         // async b32 instructions per wave per batch (= GROUPS*4)
#define NTOP    4            // top-(k-1) for k=5
#define NWAVES  (BLOCK / 32)

typedef __attribute__((address_space(1))) int  as1_int;
typedef __attribute__((address_space(3))) void as3_void;
typedef __attribute__((address_space(3))) int  as3_int;

#if defined(__AMDGCN__) && __has_builtin(__builtin_amdgcn_global_load_async_to_lds_b32)
#define USE_ASYNC_LDS 1
#else
#define USE_ASYNC_LDS 0
#endif

#if defined(__AMDGCN__) && __has_builtin(__builtin_amdgcn_s_wait_asynccnt)
#define WAIT_ASYNC(n) __builtin_amdgcn_s_wait_asynccnt(n)
#elif defined(__AMDGCN__)
#define WAIT_ASYNC(n) asm volatile("s_wait_asynccnt %0" :: "i"(n) : "memory")
#else
#define WAIT_ASYNC(n)
#endif

__device__ __forceinline__ void topn_insert(float (&t)[NTOP], float v) {
#pragma unroll
  for (int i = 0; i < NTOP; ++i) {
    float hi = fmaxf(t[i], v);
    v = fminf(t[i], v);
    t[i] = hi;
  }
}

// Merge partner's (top-4, S) into ours; S values are relative to each side's t[0].
__device__ __forceinline__ void merge_state(float (&t)[NTOP], float& S,
                                            const float (&ot)[NTOP], float oS) {
  const float m_a = t[0];
  const float m_b = ot[0];
#pragma unroll
  for (int i = 0; i < NTOP; ++i) topn_insert(t, ot[i]);
  const float mn = t[0];
  S = S * __expf(m_a - mn) + oS * __expf(m_b - mn);
}

// One workgroup per row; single streaming pass with online softmax + top-4.
// Chunks are double-buffered through LDS via CDNA5 async global->LDS (ASYNCcnt).
// Each lane async-loads exactly the 4-float quads it later reads back with
// ds_load_b128, so producer->consumer sync is wave-local (s_wait_asynccnt only).
__global__ __launch_bounds__(BLOCK) void ce_topk_row_kernel(
    const float* __restrict__ x, const long long* __restrict__ y,
    const int* __restrict__ kptr, float* __restrict__ row_loss, int M) {
  __shared__ float s_buf[2][CHUNK];          // 32 KB double buffer
  __shared__ float s_top[NWAVES][NTOP + 1];  // per-wave {top4, S}
  __shared__ float s_misc[1];                // s_y

  const int tid  = threadIdx.x;
  const int lane = tid & 31;
  const int wid  = tid >> 5;
  const int row  = blockIdx.x;
  const float* xrow = x + (size_t)row * (size_t)M;
  const int ycol = (int)y[row];
  const int nch  = (M + CHUNK - 1) / CHUNK;

  // ---- batch issue: exactly BATCH_OPS async instructions per wave (tail clamped) ----
  auto issue_batch = [&](int c, int pb) {
    const int j0 = c * CHUNK;
#pragma unroll
    for (int g = 0; g < GROUPS; ++g) {
#pragma unroll
      for (int e = 0; e < 4; ++e) {
        const int i = (tid << 2) + (g << 10) + e;  // lane-owned quad slot
        int jg = j0 + i;
        if (jg >= M) jg = M - 1;             // clamp: pads tail, slots skipped at consume
#if USE_ASYNC_LDS
        __builtin_amdgcn_global_load_async_to_lds_b32(
            (as1_int*)(uintptr_t)(xrow + jg),
            (as3_int*)(as3_void*)(&s_buf[pb][i]),
            0, 0);
#else
        s_buf[pb][i] = xrow[jg];
#endif
      }
    }
  };

  float t[NTOP];
#pragma unroll
  for (int i = 0; i < NTOP; ++i) t[i] = -3.402823466e38f;
  float S = 0.0f;

  issue_batch(0, 0);
  for (int c = 0; c < nch; ++c) {
    const int cur = c & 1;
    if (c + 1 < nch) {
      issue_batch(c + 1, cur ^ 1);           // prefetch next while we compute
      WAIT_ASYNC(BATCH_OPS);                 // <=16 outstanding -> batch c landed
    } else {
      WAIT_ASYNC(0);
    }
    const int j0 = c * CHUNK;
#pragma unroll
    for (int g = 0; g < GROUPS; ++g) {
      const int base = (tid << 2) + (g << 10);
      const float4 q = *reinterpret_cast<const float4*>(&s_buf[cur][base]);  // ds_load_b128
      const float vv[4] = {q.x, q.y, q.z, q.w};
      const int jb = j0 + base;
      const float mx = fmaxf(fmaxf(vv[0], vv[1]), fmaxf(vv[2], vv[3]));
      const bool has_label = ((unsigned)(ycol - jb) < 4u);
      const bool has_tail  = (jb + 4 > M);
      if (mx > t[NTOP - 1] || has_label || has_tail) {
        // slow path (rare): per-element insert / label record / bounds
        for (int e = 0; e < 4; ++e) {
          const int jg = jb + e;
          if (jg >= M) break;
          const float v = vv[e];
          if (jg == ycol) { s_misc[0] = v; continue; }
          if (v > t[NTOP - 1]) {
            const float m_old = t[0];
            topn_insert(t, v);
            if (t[0] > m_old) S *= __expf(m_old - t[0]);  // online-softmax rescale
          }
          S += __expf(v - t[0]);
        }
      } else {
        // fast path: 4 independent exps against loop-invariant max
        const float t0 = t[0];
        S += __expf(vv[0] - t0) + __expf(vv[1] - t0) +
             __expf(vv[2] - t0) + __expf(vv[3] - t0);
      }
    }
  }

  // ---- wave32 butterfly merge of (top4, S) ----
#pragma unroll
  for (int off = 16; off >= 1; off >>= 1) {
    float ot[NTOP];
#pragma unroll
    for (int i = 0; i < NTOP; ++i) ot[i] = __shfl_xor(t[i], off, 32);
    const float oS = __shfl_xor(S, off, 32);
    merge_state(t, S, ot, oS);
  }
  if (lane == 0) {
#pragma unroll
    for (int i = 0; i < NTOP; ++i) s_top[wid][i] = t[i];
    s_top[wid][NTOP] = S;
  }
  __syncthreads();                           // also orders the s_misc[0] write

  if (tid == 0) {
    // serial fixed-order merge of the 8 wave states -> deterministic
#pragma unroll
    for (int w = 1; w < NWAVES; ++w) {
      float ot[NTOP];
#pragma unroll
      for (int i = 0; i < NTOP; ++i) ot[i] = s_top[w][i];
      merge_state(t, S, ot, s_top[w][NTOP]);
    }
    int kk = kptr[0] - 1;                    // exclude top-(k-1) non-label entries
    if (kk < 0) kk = 0;
    if (kk > NTOP) kk = NTOP;
    const float mt = t[0];
    const float sy = s_misc[0];
    float excl = 0.0f;
    for (int i = 0; i < kk; ++i) excl += __expf(t[i] - mt);
    float sum_bottom = (S - excl) * __expf(mt - sy);
    if (sum_bottom < 0.0f) sum_bottom = 0.0f;
    row_loss[row] = __logf(sum_bottom + 1.0f);
  }
}

// Deterministic fixed-order mean of the per-row losses.
__global__ __launch_bounds__(1024) void ce_topk_mean_kernel(
    const float* __restrict__ row_loss, float* __restrict__ out, int B) {
  __shared__ float r[1024];
  const int tid = threadIdx.x;
  float v = 0.0f;
  for (int i = tid; i < B; i += 1024) v += row_loss[i];
  r[tid] = v;
  __syncthreads();
  for (int s = 512; s > 0; s >>= 1) {
    if (tid < s) r[tid] += r[tid + s];
    __syncthreads();
  }
  if (tid == 0) out[0] = r[0] / (float)B;
}

extern "C" void kernel_launch(void* const* d_in, const int* in_sizes, int n_in,
                              void* d_out, int out_size, void* d_ws, size_t ws_size,
                              hipStream_t stream) {
  (void)n_in; (void)out_size; (void)ws_size;
  const float*     x  = (const float*)d_in[0];
  const long long* y  = (const long long*)d_in[1];   // reference declares int64 labels
  const int*       kp = (const int*)d_in[2];         // low 32 bits of the scalar k
  float* out = (float*)d_out;
  float* ws  = (float*)d_ws;                         // B per-row losses

  const int B = in_sizes[1];
  const int M = in_sizes[0] / B;

  ce_topk_row_kernel<<<B, BLOCK, 0, stream>>>(x, y, kp, ws, M);
  ce_topk_mean_kernel<<<1, 1024, 0, stream>>>(ws, out, B);
}